// QueryAndGroupPN_49804440764924
// MI455X (gfx1250) — compile-verified
//
#include <hip/hip_runtime.h>

typedef __attribute__((ext_vector_type(2))) float v2f;
typedef __attribute__((ext_vector_type(4))) float v4f;
typedef __attribute__((ext_vector_type(8))) float v8f;

#define RADIUS   0.3f
#define R2       (RADIUS * RADIUS)
#define NSAMPLE  64
#define B_       16
#define N_       8192
#define P_       1024
#define C_       256
#define OUTC     (3 + C_)     // 259
#define CHUNK    512          // points staged per iteration
#define CPB      16           // centers per block (one WMMA M-tile)
#define THREADS_A 256

// ---------------------------------------------------------------------------
// Kernel A: ball query. One block = 16 centers of one batch.
//   - WMMA f32 16x16x4 computes cross = -2 * <center, point> for 16x16 tiles
//   - ballot-based in-order append reproduces CUDA ball_query semantics
//   - early exit once all 16 centers have 64 hits
// ---------------------------------------------------------------------------
__global__ __launch_bounds__(THREADS_A) void ballquery_wmma_kernel(
    const float* __restrict__ xyz,      // [B, N, 3]
    const float* __restrict__ new_xyz,  // [B, P, 3]
    int* __restrict__ idx_out)          // [B, P, NSAMPLE]
{
  __shared__ float s_cx[CPB], s_cy[CPB], s_cz[CPB], s_cn[CPB];
  __shared__ float s_px[CHUNK], s_py[CHUNK], s_pz[CHUNK], s_pn[CHUNK];
  __shared__ float s_cross[CPB * CHUNK];   // 32 KB: -2*dot for 16 centers x 512 pts
  __shared__ int   s_count[CPB];
  __shared__ int   s_first[CPB];
  __shared__ int   s_alldone;

  const int t    = threadIdx.x;
  const int lane = t & 31;
  const int wave = t >> 5;
  const int b    = blockIdx.x / (P_ / CPB);
  const int p0   = (blockIdx.x % (P_ / CPB)) * CPB;

  if (t < CPB) {
    const float* c = new_xyz + ((size_t)b * P_ + p0 + t) * 3;
    float x = c[0], y = c[1], z = c[2];
    s_cx[t] = x; s_cy[t] = y; s_cz[t] = z;
    s_cn[t] = x * x + y * y + z * z;
    s_count[t] = 0; s_first[t] = 0;
  }
  if (t == 0) s_alldone = 0;
  __syncthreads();

  // A operand (16x4 f32, M=centers, K=xyz0) scaled by -2 so D = -2*<c,p>.
  // ISA layout: lanes 0-15 hold (K0,K1) of M=lane; lanes 16-31 hold (K2,K3).
  v2f Aop;
  {
    int c = lane & 15;
    Aop.x = (lane < 16) ? (-2.0f * s_cx[c]) : (-2.0f * s_cz[c]);
    Aop.y = (lane < 16) ? (-2.0f * s_cy[c]) : 0.0f;
  }

  const float* xb = xyz + (size_t)b * N_ * 3;

  for (int base = 0; base < N_; base += CHUNK) {
    // ---- stage point chunk + norms into LDS -----------------------------
    for (int i = t; i < CHUNK; i += THREADS_A) {
      const float* g = xb + (size_t)(base + i) * 3;
      float x = g[0], y = g[1], z = g[2];
      s_px[i] = x; s_py[i] = y; s_pz[i] = z;
      s_pn[i] = x * x + y * y + z * z;
    }
    __syncthreads();

    // ---- cross terms via V_WMMA_F32_16X16X4_F32 (32 tiles / 8 waves) ----
    for (int tile = wave; tile < CHUNK / 16; tile += 8) {
      const int tb = tile * 16;
      const int q  = tb + (lane & 15);
      // B operand (4x16 f32, K=xyz0, N=points): mirror of A layout.
      v2f Bop;
      Bop.x = (lane < 16) ? s_px[q] : s_pz[q];
      Bop.y = (lane < 16) ? s_py[q] : 0.0f;
      v8f acc = {};
      acc = __builtin_amdgcn_wmma_f32_16x16x4_f32(
          false, Aop, false, Bop, (short)0, acc, false, false);
      // D layout: VGPR r -> row M=r (lanes 0-15) / M=r+8 (lanes 16-31)
      const int col   = tb + (lane & 15);
      const int rbase = (lane < 16) ? 0 : 8;
#pragma unroll
      for (int r = 0; r < 8; ++r)
        s_cross[(rbase + r) * CHUNK + col] = acc[r];
    }
    __syncthreads();

    // ---- in-order selection: wave w owns centers 2w, 2w+1 ----------------
    for (int ci = 0; ci < 2; ++ci) {
      const int c = wave * 2 + ci;
      int count = s_count[c];
      if (count < NSAMPLE) {
        const float cn = s_cn[c];
        int* dst = idx_out + ((size_t)(b * P_ + p0 + c)) * NSAMPLE;
        for (int step = 0; step < CHUNK / 32 && count < NSAMPLE; ++step) {
          const int nl = step * 32 + lane;
          const float d2 = cn + s_pn[nl] + s_cross[c * CHUNK + nl];
          const bool in = d2 < R2;
          const unsigned mask = __builtin_amdgcn_ballot_w32(in);
          if (in) {
            const int pos = count + __popc(mask & ((1u << lane) - 1u));
            if (pos < NSAMPLE) dst[pos] = base + nl;
            if (pos == 0) s_first[c] = base + nl;
          }
          count += __popc(mask);
        }
        if (lane == 0) s_count[c] = count;
      }
    }
    __syncthreads();

    if (t == 0) {
      int done = 1;
      for (int c = 0; c < CPB; ++c) done &= (s_count[c] >= NSAMPLE);
      s_alldone = done;
    }
    __syncthreads();
    if (s_alldone) break;   // uniform across block
  }

  // ---- fill pass: pad empty slots with first hit (or 0 if none) ----------
  for (int ci = 0; ci < 2; ++ci) {
    const int c = wave * 2 + ci;
    const int cnt = min(s_count[c], NSAMPLE);
    const int first = (s_count[c] > 0) ? s_first[c] : 0;
    int* dst = idx_out + ((size_t)(b * P_ + p0 + c)) * NSAMPLE;
    for (int s = cnt + lane; s < NSAMPLE; s += 32) dst[s] = first;
  }
}

// ---------------------------------------------------------------------------
// Kernel B: gather + concat. One block per (b, p) center.
// Streaming non-temporal stores keep the 134 MB features array L2-resident.
// ---------------------------------------------------------------------------
__global__ __launch_bounds__(256) void gather_group_kernel(
    const float* __restrict__ xyz,       // [B, N, 3]
    const float* __restrict__ new_xyz,   // [B, P, 3]
    const float* __restrict__ feat,      // [B, C, N]
    const int*   __restrict__ idx,       // [B, P, NSAMPLE]
    float* __restrict__ out)             // [B, OUTC, P, NSAMPLE]
{
  __shared__ int   sidx[NSAMPLE];
  __shared__ float scen[3];

  const int t = threadIdx.x;
  const int b = blockIdx.x / P_;
  const int p = blockIdx.x % P_;
  const size_t bp = (size_t)b * P_ + p;

  if (t < NSAMPLE) sidx[t] = idx[bp * NSAMPLE + t];
  if (t < 3)       scen[t] = new_xyz[bp * 3 + t];
  __syncthreads();

  // xyz channels 0..2: (xyz[idx] - center) / RADIUS
  if (t < 3 * NSAMPLE) {
    const int ch = t >> 6, k = t & 63;
    const float v =
        (xyz[((size_t)b * N_ + sidx[k]) * 3 + ch] - scen[ch]) * (1.0f / RADIUS);
    __builtin_nontemporal_store(
        v, &out[(((size_t)b * OUTC + ch) * P_ + p) * NSAMPLE + k]);
  }

  // feature channels: 256 ch x 64 k, 4 k per thread (b128 NT stores)
  const float* fb = feat + (size_t)b * C_ * N_;
#pragma unroll 4
  for (int it = 0; it < (C_ * (NSAMPLE / 4)) / 256; ++it) {  // 16 iterations
    const int item = it * 256 + t;
    const int ch   = item >> 4;          // 16 float4 per channel row
    const int k4   = (item & 15) << 2;
    const float* frow = fb + (size_t)ch * N_;
    v4f v;
    v.x = frow[sidx[k4 + 0]];
    v.y = frow[sidx[k4 + 1]];
    v.z = frow[sidx[k4 + 2]];
    v.w = frow[sidx[k4 + 3]];
    float* o = &out[(((size_t)b * OUTC + 3 + ch) * P_ + p) * NSAMPLE + k4];
    __builtin_nontemporal_store(v, (v4f*)o);
  }
}

extern "C" void kernel_launch(void* const* d_in, const int* in_sizes, int n_in,
                              void* d_out, int out_size, void* d_ws, size_t ws_size,
                              hipStream_t stream) {
  (void)in_sizes; (void)n_in; (void)out_size; (void)ws_size;
  const float* xyz      = (const float*)d_in[0];
  const float* new_xyz  = (const float*)d_in[1];
  const float* features = (const float*)d_in[2];
  float* out = (float*)d_out;
  int* idx_ws = (int*)d_ws;   // needs B*P*NSAMPLE*4 = 4 MB

  ballquery_wmma_kernel<<<dim3(B_ * (P_ / CPB)), THREADS_A, 0, stream>>>(
      xyz, new_xyz, idx_ws);
  gather_group_kernel<<<dim3(B_ * P_), 256, 0, stream>>>(
      xyz, new_xyz, features, idx_ws, out);
}